// SparseMaskEncoding_66846870995029
// MI455X (gfx1250) — compile-verified
//
#include <hip/hip_runtime.h>
#include <math.h>

typedef float v2f __attribute__((ext_vector_type(2)));
typedef float v8f __attribute__((ext_vector_type(8)));

#define WMMA_F32(a, b, c) \
  __builtin_amdgcn_wmma_f32_16x16x4_f32(false, (a), false, (b), (short)0, (c), false, false)

#define N_SAMPLES 32768
#define N_FEAT    1024
#define N_CODES   128
#define MAX_ITER  80
#define ALPHA     0.2f

// ---------------------------------------------------------------------------
// Kernel 1: L = ||A||_F^2  (single block reduction)
// ---------------------------------------------------------------------------
__global__ void k_lipschitz(const float* __restrict__ D, float* __restrict__ Lp) {
  __shared__ float red[256];
  float s = 0.0f;
  for (int i = threadIdx.x; i < N_CODES * N_FEAT; i += 256) {
    float v = D[i];
    s += v * v;
  }
  red[threadIdx.x] = s;
  __syncthreads();
  for (int off = 128; off > 0; off >>= 1) {
    if ((int)threadIdx.x < off) red[threadIdx.x] += red[threadIdx.x + off];
    __syncthreads();
  }
  if (threadIdx.x == 0) Lp[0] = red[0];
}

// ---------------------------------------------------------------------------
// Kernel 2: M = I - (A A^T)/L   (128x128, symmetric; tiny one-off)
// ---------------------------------------------------------------------------
__global__ void k_gram(const float* __restrict__ D, const float* __restrict__ Lp,
                       float* __restrict__ M) {
  int t = blockIdx.x * blockDim.x + threadIdx.x;  // 0..16383
  int i = t >> 7, j = t & 127;
  const float4* ai = (const float4*)(D + (size_t)i * N_FEAT);
  const float4* aj = (const float4*)(D + (size_t)j * N_FEAT);
  float s = 0.0f;
  for (int k = 0; k < N_FEAT / 4; ++k) {
    float4 x = ai[k], y = aj[k];
    s += x.x * y.x + x.y * y.y + x.z * y.z + x.w * y.w;
  }
  float invL = 1.0f / Lp[0];
  M[t] = ((i == j) ? 1.0f : 0.0f) - s * invL;
}

// ---------------------------------------------------------------------------
// Kernel 3: B = (X A^T) / L  -> written into d_out (same shape as codes).
// WMMA f32 16x16x4, K streamed over 1024 features. Each wave: 16 rows.
// ---------------------------------------------------------------------------
__global__ __launch_bounds__(128) void k_bmat(const float* __restrict__ X,
                                              const float* __restrict__ D,
                                              const float* __restrict__ Lp,
                                              float* __restrict__ B) {
  const int lane = threadIdx.x & 31;
  const int wave = threadIdx.x >> 5;
  const int half = lane >> 4;   // 0: K lo pair, 1: K hi pair
  const int ln   = lane & 15;
  const int row0 = blockIdx.x * 64 + wave * 16;
  const float invL = 1.0f / Lp[0];

  v8f acc[8];
#pragma unroll
  for (int j = 0; j < 8; ++j) acc[j] = (v8f){};

  const float* xrow = X + (size_t)(row0 + ln) * N_FEAT + 2 * half;
  for (int k0 = 0; k0 < N_FEAT; k0 += 4) {
    // A operand: X rows (lane ln holds row ln, K = k0+2*half+{0,1})
    v2f a = *(const v2f*)(xrow + k0);
#pragma unroll
    for (int j = 0; j < 8; ++j) {
      // B operand: A^T tile -> Bop[k][n] = D[16j+n][k0+k]; contiguous per lane
      v2f b = *(const v2f*)(D + (size_t)(16 * j + ln) * N_FEAT + k0 + 2 * half);
      acc[j] = WMMA_F32(a, b, acc[j]);
    }
  }
  // C layout: VGPR i -> row i + 8*half, col 16*j + ln
#pragma unroll
  for (int j = 0; j < 8; ++j)
#pragma unroll
    for (int i = 0; i < 8; ++i)
      B[(size_t)(row0 + i + 8 * half) * N_CODES + 16 * j + ln] = acc[j][i] * invL;
}

// ---------------------------------------------------------------------------
// Kernel 4: 80 FISTA iterations, fully resident.
//   z_grad = z @ M + B  (M symmetric, in LDS; z staged per-wave in LDS)
//   x = soft_thresh(z_grad, alpha/L);  z = x + mom*(x - x_old)
// Reads B from `out`, finally overwrites `out` with x.
// ---------------------------------------------------------------------------
__global__ __launch_bounds__(128) void k_fista(const float* __restrict__ Mg,
                                               const float* __restrict__ Lp,
                                               float* __restrict__ out) {
  __shared__ float Ml[N_CODES * N_CODES];   // 64 KB
  __shared__ float zb[4][16 * N_CODES];     // 4 waves x 8 KB
  const int lane = threadIdx.x & 31;
  const int wave = threadIdx.x >> 5;
  const int half = lane >> 4;
  const int ln   = lane & 15;
  const int row0 = blockIdx.x * 64 + wave * 16;
  const float thr = ALPHA / Lp[0];

  for (int i = threadIdx.x; i < N_CODES * N_CODES; i += 128) Ml[i] = Mg[i];
  float* z = zb[wave];
  for (int i = lane; i < 16 * N_CODES; i += 32) z[i] = 0.0f;
  __syncthreads();

  // Load B tile (currently stored in `out`) into C-fragment layout; x0 = 0.
  v8f bf[8], xf[8];
#pragma unroll
  for (int j = 0; j < 8; ++j) {
    v8f t = (v8f){};
#pragma unroll
    for (int i = 0; i < 8; ++i)
      t[i] = out[(size_t)(row0 + i + 8 * half) * N_CODES + 16 * j + ln];
    bf[j] = t;
    xf[j] = (v8f){};
  }

  float t = 1.0f;
  for (int it = 0; it < MAX_ITER; ++it) {
    v8f acc[8];
#pragma unroll
    for (int j = 0; j < 8; ++j) acc[j] = bf[j];

    // z @ M : 32 K-steps x 8 N-tiles of v_wmma_f32_16x16x4_f32
    for (int k0 = 0; k0 < N_CODES; k0 += 4) {
      v2f a = *(const v2f*)&z[ln * N_CODES + k0 + 2 * half];
#pragma unroll
      for (int j = 0; j < 8; ++j) {
        // Bop[k][n] = M[k0+k][16j+n] = M[16j+n][k0+k]  (M symmetric)
        v2f b = *(const v2f*)&Ml[(16 * j + ln) * N_CODES + k0 + 2 * half];
        acc[j] = WMMA_F32(a, b, acc[j]);
      }
    }

    float tn  = 0.5f * (1.0f + sqrtf(1.0f + 4.0f * t * t));
    float mom = (t - 1.0f) / tn;
    t = tn;

#pragma unroll
    for (int j = 0; j < 8; ++j) {
#pragma unroll
      for (int i = 0; i < 8; ++i) {
        float zv = acc[j][i];
        float mag = fabsf(zv) - thr;
        float xn  = (mag > 0.0f) ? copysignf(mag, zv) : 0.0f;
        float zn  = xn + mom * (xn - xf[j][i]);
        xf[j][i]  = xn;
        z[(i + 8 * half) * N_CODES + 16 * j + ln] = zn;  // per-wave region
      }
    }
  }

#pragma unroll
  for (int j = 0; j < 8; ++j)
#pragma unroll
    for (int i = 0; i < 8; ++i)
      out[(size_t)(row0 + i + 8 * half) * N_CODES + 16 * j + ln] = xf[j][i];
}

// ---------------------------------------------------------------------------
extern "C" void kernel_launch(void* const* d_in, const int* in_sizes, int n_in,
                              void* d_out, int out_size, void* d_ws, size_t ws_size,
                              hipStream_t stream) {
  const float* X = (const float*)d_in[0];   // (32768, 1024)
  const float* D = (const float*)d_in[1];   // (128, 1024)
  float* out = (float*)d_out;               // (32768, 128)
  float* wsf = (float*)d_ws;
  float* Lp = wsf;                          // 1 float (aligned slot of 16)
  float* M  = wsf + 16;                     // 128*128 floats

  k_lipschitz<<<1, 256, 0, stream>>>(D, Lp);
  k_gram<<<64, 256, 0, stream>>>(D, Lp, M);
  k_bmat<<<N_SAMPLES / 64, 128, 0, stream>>>(X, D, Lp, out);
  k_fista<<<N_SAMPLES / 64, 128, 0, stream>>>(M, Lp, out);
}